// PT_LSTM_52355651338920
// MI455X (gfx1250) — compile-verified
//
#include <hip/hip_runtime.h>
#include <cstdint>
#include <cstddef>

// Problem constants (fixed by the reference).
#define MEMSZ 4096   // M: hidden size (cols of every weight matrix)
#define INSZ  2048   // I: input size
#define KSPLIT 32    // split-K slices for the GEMVs (deterministic two-stage reduce)

typedef __attribute__((ext_vector_type(2))) float v2f;
typedef __attribute__((ext_vector_type(8))) float v8f;

// ---------------------------------------------------------------------------
// z[g*M + j] = bias_g[j]   (gate order: inpgate, forgetgate, outgate, inp)
// ---------------------------------------------------------------------------
__global__ void lstm_init_z(float* __restrict__ z,
                            const float* __restrict__ b_ig,
                            const float* __restrict__ b_fg,
                            const float* __restrict__ b_og,
                            const float* __restrict__ b_in) {
  int j = blockIdx.x * blockDim.x + threadIdx.x;      // 0..4095
  z[j]           = b_ig[j];
  z[MEMSZ   + j] = b_fg[j];
  z[2*MEMSZ + j] = b_og[j];
  z[3*MEMSZ + j] = b_in[j];
}

// ---------------------------------------------------------------------------
// Split-K GEMV partial: zpart[s][j] = sum_{i in slice s} v[i] * W[i][j].
// Row-major W [rows x 4096]; each thread owns 4 consecutive columns
// (GLOBAL_LOAD_B128 per row, fully coalesced across the wave).
// Grid: (4096/4/256 = 4, KSPLIT).  Deterministic (no atomics).
// Recurrent matrices pass skip_if_reset=1 and skip all traffic when reset.
// ---------------------------------------------------------------------------
__global__ void lstm_gemv_part(const float* __restrict__ W,
                               const float* __restrict__ v,
                               float* __restrict__ zpart,
                               int rows_per_slice,
                               const int* __restrict__ is_reset,
                               int skip_if_reset) {
  if (skip_if_reset && (*is_reset != 0)) return;      // uniform branch, whole block exits
  const int j4 = blockIdx.x * blockDim.x + threadIdx.x;   // float4 column index 0..1023
  const int s  = blockIdx.y;
  const int i0 = s * rows_per_slice;
  const float4* __restrict__ Wp = reinterpret_cast<const float4*>(W) + (size_t)i0 * (MEMSZ/4) + j4;
  float4 acc = make_float4(0.f, 0.f, 0.f, 0.f);
  #pragma unroll 4
  for (int i = 0; i < rows_per_slice; ++i) {
    const float vv = v[i0 + i];                       // uniform across wave
    const float4 w = Wp[(size_t)i * (MEMSZ/4)];
    acc.x = fmaf(vv, w.x, acc.x);
    acc.y = fmaf(vv, w.y, acc.y);
    acc.z = fmaf(vv, w.z, acc.z);
    acc.w = fmaf(vv, w.w, acc.w);
  }
  reinterpret_cast<float4*>(zpart)[(size_t)s * (MEMSZ/4) + j4] = acc;
}

// ---------------------------------------------------------------------------
// z[j] += sum_{s=0..KSPLIT-1} zpart[s][j], fixed order -> deterministic.
// Must mirror the producer's skip so stale zpart is never consumed.
// ---------------------------------------------------------------------------
__global__ void lstm_reduce_add(const float* __restrict__ zpart,
                                float* __restrict__ z,
                                const int* __restrict__ is_reset,
                                int skip_if_reset) {
  if (skip_if_reset && (*is_reset != 0)) return;
  int j = blockIdx.x * blockDim.x + threadIdx.x;      // 0..4095
  float acc = 0.f;
  #pragma unroll
  for (int s = 0; s < KSPLIT; ++s) acc += zpart[(size_t)s * MEMSZ + j];
  z[j] += acc;
}

// ---------------------------------------------------------------------------
// Gate nonlinearity + cell/ hidden update:
//   ig=s(z0) fg=s(z1) og=s(z2) ; c_t = ig*tanh(z3) + fg*c_prev ; h_t = og*tanh(c_t)
// ---------------------------------------------------------------------------
__global__ void lstm_gates(const float* __restrict__ z,
                           const float* __restrict__ c0,
                           const int* __restrict__ is_reset,
                           float* __restrict__ ht) {
  int j = blockIdx.x * blockDim.x + threadIdx.x;      // 0..4095
  const float zi = z[j];
  const float zf = z[MEMSZ   + j];
  const float zo = z[2*MEMSZ + j];
  const float zc = z[3*MEMSZ + j];
  const float cprev = (*is_reset != 0) ? 0.f : c0[j];
  const float ig = 1.f / (1.f + expf(-zi));
  const float fg = 1.f / (1.f + expf(-zf));
  const float og = 1.f / (1.f + expf(-zo));
  const float ct = ig * tanhf(zc) + fg * cprev;
  ht[j] = og * tanhf(ct);
}

// ---------------------------------------------------------------------------
// Rank-1 outer product out[i][j] = w_hid[i] * h_t[j] via V_WMMA_F32_16X16X4_F32.
// Permutation-proof operand build: A[m][k] = w[i0+m] for ALL k (A K-map is
// documented but irrelevant here), B[k][n] = h[j0+n]/4 for ALL k, so
// D[m][n] = sum_k w*h/4 = w[m]*h[n] exactly (x0.25 and the 4x sum are exact
// power-of-two ops in fp32) under ANY (vgpr,half)->K slot assignment.
// One wave per 16x16 output tile; 8 waves (256 thr) per block. EXEC all-ones.
// Store uses the documented C/D layout: lane half selects M in {r, 8+r}.
// ---------------------------------------------------------------------------
__global__ void __launch_bounds__(256)
lstm_outer_wmma(const float* __restrict__ whid,
                const float* __restrict__ ht,
                float* __restrict__ out) {
  const int lane = threadIdx.x;                 // 0..31
  const int wv   = threadIdx.y;                 // 0..7
  const int ti   = blockIdx.y;                  // row tile  0..255
  const int tj   = blockIdx.x * 8 + wv;         // col tile  0..255
  const int l15  = lane & 15;
  const int half = lane >> 4;

  const float a = whid[ti * 16 + l15];          // replicate across all K slots
  const float b = 0.25f * ht[tj * 16 + l15];    // replicate across all K slots

  v2f A; A[0] = a; A[1] = a;
  v2f B; B[0] = b; B[1] = b;
  v8f C = {};
  v8f D = __builtin_amdgcn_wmma_f32_16x16x4_f32(
      /*neg_a=*/false, A, /*neg_b=*/false, B,
      /*c_mod=*/(short)0, C, /*reuse_a=*/false, /*reuse_b=*/false);

  const size_t base = (size_t)(ti * 16 + half * 8) * MEMSZ + (size_t)tj * 16 + l15;
  #pragma unroll
  for (int r = 0; r < 8; ++r) out[base + (size_t)r * MEMSZ] = D[r];
}

// ---------------------------------------------------------------------------
// Host-side launcher. Input order follows setup_inputs():
//  0 x  1 h0  2 c0
//  3 w_inpgate  4 w_rec_inpgate  5 w_mem_inpgate
//  6 w_inp      7 w_rec_inp
//  8 w_forgetgate  9 w_rec_forgetgate 10 w_mem_forgetgate
// 11 w_outgate    12 w_rec_outgate    13 w_mem_outgate
// 14 w_hid_out
// 15 b_inpgate 16 b_inp 17 b_forgetgate 18 b_outgate
// 19 is_reset (int, size 1)
// ---------------------------------------------------------------------------
extern "C" void kernel_launch(void* const* d_in, const int* in_sizes, int n_in,
                              void* d_out, int out_size, void* d_ws, size_t ws_size,
                              hipStream_t stream) {
  const float* x  = (const float*)d_in[0];
  const float* h0 = (const float*)d_in[1];
  const float* c0 = (const float*)d_in[2];
  const float* w_ig  = (const float*)d_in[3];
  const float* w_rig = (const float*)d_in[4];
  const float* w_mig = (const float*)d_in[5];
  const float* w_in  = (const float*)d_in[6];
  const float* w_rin = (const float*)d_in[7];
  const float* w_fg  = (const float*)d_in[8];
  const float* w_rfg = (const float*)d_in[9];
  const float* w_mfg = (const float*)d_in[10];
  const float* w_og  = (const float*)d_in[11];
  const float* w_rog = (const float*)d_in[12];
  const float* w_mog = (const float*)d_in[13];
  const float* w_hid = (const float*)d_in[14];
  const float* b_ig  = (const float*)d_in[15];
  const float* b_in  = (const float*)d_in[16];
  const float* b_fg  = (const float*)d_in[17];
  const float* b_og  = (const float*)d_in[18];
  const int*   is_rst = (const int*)d_in[19];

  float* out = (float*)d_out;                   // [4096,4096] row-major

  // Workspace layout: z[4*M] | ht[M] | zpart[KSPLIT*M]  => ~0.6 MB
  float* z     = (float*)d_ws;
  float* ht    = z + 4 * MEMSZ;
  float* zpart = ht + MEMSZ;

  const dim3 blk256(256);
  const dim3 grd16(MEMSZ / 256);                               // 4096 threads
  const dim3 gemv_grid(MEMSZ / 4 / 256, KSPLIT);               // (4, 32)
  const int rows_x   = INSZ  / KSPLIT;                         // 64
  const int rows_rec = MEMSZ / KSPLIT;                         // 128

  // z = bias
  lstm_init_z<<<grd16, blk256, 0, stream>>>(z, b_ig, b_fg, b_og, b_in);

  // z += x @ Wx  (4 matrices, 128 MB total — the dominant read stream)
  const float* Wx[4]  = {w_ig, w_fg, w_og, w_in};
  for (int g = 0; g < 4; ++g) {
    lstm_gemv_part<<<gemv_grid, blk256, 0, stream>>>(Wx[g], x, zpart, rows_x, is_rst, 0);
    lstm_reduce_add<<<grd16, blk256, 0, stream>>>(zpart, z + g * MEMSZ, is_rst, 0);
  }

  // z += h0 @ Wh (4 matrices) and z[0:3M] += c0 @ Wc (3 matrices);
  // both are exactly zero when is_reset -> kernels early-exit, no traffic.
  const float* Wh[4] = {w_rig, w_rfg, w_rog, w_rin};
  for (int g = 0; g < 4; ++g) {
    lstm_gemv_part<<<gemv_grid, blk256, 0, stream>>>(Wh[g], h0, zpart, rows_rec, is_rst, 1);
    lstm_reduce_add<<<grd16, blk256, 0, stream>>>(zpart, z + g * MEMSZ, is_rst, 1);
  }
  const float* Wc[3] = {w_mig, w_mfg, w_mog};
  for (int g = 0; g < 3; ++g) {
    lstm_gemv_part<<<gemv_grid, blk256, 0, stream>>>(Wc[g], c0, zpart, rows_rec, is_rst, 1);
    lstm_reduce_add<<<grd16, blk256, 0, stream>>>(zpart, z + g * MEMSZ, is_rst, 1);
  }

  // Nonlinearities -> h_t
  lstm_gates<<<grd16, blk256, 0, stream>>>(z, c0, is_rst, ht);

  // out = w_hid_out (4096x1) @ h_t (1x4096), 67 MB store stream via WMMA tiles.
  lstm_outer_wmma<<<dim3(256 / 8, 256), dim3(32, 8), 0, stream>>>(w_hid, ht, out);
}